// DGAnet_16381005267294
// MI455X (gfx1250) — compile-verified
//
#include <hip/hip_runtime.h>

#define N_NODES 100000
#define N_EDGES 1000000
#define IN_C 64
#define HID_C 64
#define OUT_C 32
#define N_TILES_ROWS (N_NODES / 16)   // 6250, exact

typedef __attribute__((ext_vector_type(2))) float v2f;
typedef __attribute__((ext_vector_type(8))) float v8f;

// ---------------------------------------------------------------- zero
__global__ void zero_kernel(float* __restrict__ p, int n) {
    int i = blockIdx.x * blockDim.x + threadIdx.x;
    int stride = gridDim.x * blockDim.x;
    for (; i < n; i += stride) p[i] = 0.0f;
}

// ---------------------------------------------------------------- scatter
// 16 threads per edge; thread handles 4 channels (float4 gather, 4 atomics).
__global__ void scatter_kernel(const float* __restrict__ feat,
                               const int* __restrict__ ei,
                               float* __restrict__ aggr,
                               float* __restrict__ deg,
                               int add_deg) {
    unsigned gid = blockIdx.x * blockDim.x + threadIdx.x;   // < 16M
    unsigned e    = gid >> 4;
    unsigned part = gid & 15;
    if (e >= N_EDGES) return;
    int src = ei[e];
    int dst = ei[N_EDGES + e];
    const float4 v = ((const float4*)(feat + (size_t)src * 64))[part];
    float* dptr = aggr + (size_t)dst * 64 + part * 4;
    unsafeAtomicAdd(dptr + 0, v.x);
    unsafeAtomicAdd(dptr + 1, v.y);
    unsafeAtomicAdd(dptr + 2, v.z);
    unsafeAtomicAdd(dptr + 3, v.w);
    if (add_deg && part == 0) unsafeAtomicAdd(deg + dst, 1.0f);
}

// ---------------------------------------------------------------- SAGE GEMM
// out[16 rows x NC] = (aggr/deg) @ Wl + x @ Wr + bias  (optional ReLU)
// One wave per 16-row tile; K = 64 via 16 steps of V_WMMA_F32_16X16X4_F32.
template <int NTILES, bool RELU>
__global__ __launch_bounds__(256) void sage_gemm_kernel(
    const float* __restrict__ aggr, const float* __restrict__ deg,
    const float* __restrict__ xin,  const float* __restrict__ Wl,
    const float* __restrict__ bias, const float* __restrict__ Wr,
    float* __restrict__ out) {
    constexpr int NC = NTILES * 16;
    __shared__ float sWl[64 * NC];
    __shared__ float sWr[64 * NC];
    __shared__ float sB[NC];

    for (int i = threadIdx.x; i < 64 * NC; i += 256) {
        sWl[i] = Wl[i];
        sWr[i] = Wr[i];
    }
    if ((int)threadIdx.x < NC) sB[threadIdx.x] = bias[threadIdx.x];
    __syncthreads();

    const int wave = threadIdx.x >> 5;
    const int lane = threadIdx.x & 31;
    const int tile = blockIdx.x * 8 + wave;
    if (tile >= N_TILES_ROWS) return;   // wave-uniform: EXEC stays all-ones

    const int lane15 = lane & 15;
    const int kgrp   = lane >> 4;       // 0 or 1
    const int row    = tile * 16 + lane15;

    const float inv = 1.0f / fmaxf(deg[row], 1.0f);
    const float* __restrict__ arow = aggr + (size_t)row * 64;
    const float* __restrict__ xrow = xin  + (size_t)row * 64;

    v8f acc[NTILES] = {};

    for (int ks = 0; ks < 16; ++ks) {
        const int k0 = ks * 4 + kgrp * 2;
        v2f aA, aX;
        aA.x = arow[k0] * inv;
        aA.y = arow[k0 + 1] * inv;
        aX.x = xrow[k0];
        aX.y = xrow[k0 + 1];
#pragma unroll
        for (int nt = 0; nt < NTILES; ++nt) {
            const int col = nt * 16 + lane15;
            v2f bL, bR;
            bL.x = sWl[k0 * NC + col];
            bL.y = sWl[(k0 + 1) * NC + col];
            bR.x = sWr[k0 * NC + col];
            bR.y = sWr[(k0 + 1) * NC + col];
            acc[nt] = __builtin_amdgcn_wmma_f32_16x16x4_f32(
                false, aA, false, bL, (short)0, acc[nt], false, false);
            acc[nt] = __builtin_amdgcn_wmma_f32_16x16x4_f32(
                false, aX, false, bR, (short)0, acc[nt], false, false);
        }
    }

#pragma unroll
    for (int nt = 0; nt < NTILES; ++nt) {
        const int col = nt * 16 + lane15;
        const float b = sB[col];
#pragma unroll
        for (int v = 0; v < 8; ++v) {
            const int orow = tile * 16 + v + 8 * kgrp;
            float val = acc[nt][v] + b;
            if (RELU) val = fmaxf(val, 0.0f);
            out[(size_t)orow * NC + col] = val;
        }
    }
}

// ---------------------------------------------------------------- launch
extern "C" void kernel_launch(void* const* d_in, const int* in_sizes, int n_in,
                              void* d_out, int out_size, void* d_ws, size_t ws_size,
                              hipStream_t stream) {
    const float* x   = (const float*)d_in[0];
    const int*   ei  = (const int*)d_in[1];
    const float* W1l = (const float*)d_in[2];
    const float* b1  = (const float*)d_in[3];
    const float* W1r = (const float*)d_in[4];
    const float* W2l = (const float*)d_in[5];
    const float* b2  = (const float*)d_in[6];
    const float* W2r = (const float*)d_in[7];
    float* out = (float*)d_out;

    float* ws   = (float*)d_ws;
    float* deg  = ws;                               // [N_NODES]
    float* aggr = ws + N_NODES;                     // [N_NODES*64], 16B-aligned
    float* h    = aggr + (size_t)N_NODES * 64;      // [N_NODES*64]

    const int scatter_blocks = (N_EDGES * 16) / 256;          // 62500
    const int gemm_blocks    = (N_TILES_ROWS + 7) / 8;        // 782

    // Layer 1
    zero_kernel<<<4096, 256, 0, stream>>>(deg, N_NODES + N_NODES * 64);
    scatter_kernel<<<scatter_blocks, 256, 0, stream>>>(x, ei, aggr, deg, 1);
    sage_gemm_kernel<4, true><<<gemm_blocks, 256, 0, stream>>>(
        aggr, deg, x, W1l, b1, W1r, h);

    // Layer 2
    zero_kernel<<<4096, 256, 0, stream>>>(aggr, N_NODES * 64);
    scatter_kernel<<<scatter_blocks, 256, 0, stream>>>(h, ei, aggr, nullptr, 0);
    sage_gemm_kernel<2, false><<<gemm_blocks, 256, 0, stream>>>(
        aggr, deg, h, W2l, b2, W2r, out);
}